// QuasiMetric_23596550324390
// MI455X (gfx1250) — compile-verified
//
#include <hip/hip_runtime.h>

// IQE quasimetric distance: per row, measure of the union of intervals
// [x_i, y_i] for i where x_i < y_i.  One wave32 per row: stage x,y rows to
// LDS with CDNA5 async global->LDS copies (inline asm, ASYNCcnt-tracked),
// blocked bitonic sort of the 256 tagged endpoints (8 elements/lane,
// shfl_xor for lane-crossing strides), wave prefix-scan of +/-1 tags, then
// the telescoped segment sum.

#define ZDIM   128
#define N2     256          // 2*ZDIM endpoints per row
#define EPL    8            // endpoints per lane (N2 / 32)
#define WAVES  8            // waves (rows) per block
#define BLK    (WAVES * 32)

// CDNA5 async copy: 16B per lane from global g into LDS at byte offset lds_off.
// lds_off is the AS3 byte offset: on AMDGPU the flat pointer to __shared__
// data is {shared_aperture_hi32, lds_offset_lo32} (ISA 10.2), so truncating
// the generic pointer gives the value VDST expects.
__device__ __forceinline__ void stage_b128_async(const float* g, float* l) {
  unsigned lds_off = (unsigned)(unsigned long long)l;
  asm volatile("global_load_async_to_lds_b128 %0, %1, off"
               :: "v"(lds_off), "v"(g)
               : "memory");
}

__device__ __forceinline__ void wait_async_zero() {
#if __has_builtin(__builtin_amdgcn_s_wait_asynccnt)
  __builtin_amdgcn_s_wait_asynccnt(0);
  asm volatile("" ::: "memory");
#else
  asm volatile("s_wait_asynccnt 0" ::: "memory");
#endif
}

__global__ __launch_bounds__(BLK) void iqe_union_len_kernel(
    const float* __restrict__ x, const float* __restrict__ y,
    float* __restrict__ out, int nrows) {
  __shared__ float smem[WAVES * N2];   // 8 KB / block

  const int lane = threadIdx.x & 31;
  const int wave = threadIdx.x >> 5;
  const long row = (long)blockIdx.x * WAVES + wave;
  if (row >= nrows) return;            // uniform per wave

  float* ws = &smem[wave * N2];        // [0..127]=x row, [128..255]=y row

  // ---- async global -> LDS staging (ASYNCcnt), 16B per lane per row ----
  const float* xrow = x + row * ZDIM;
  const float* yrow = y + row * ZDIM;
  stage_b128_async(xrow + lane * 4, ws + lane * 4);
  stage_b128_async(yrow + lane * 4, ws + ZDIM + lane * 4);
  wait_async_zero();                   // only this wave's own copies

  // ---- gather 8 own endpoints + 8 counterparts, build keys and tags ----
  // global element index i = lane*8 + r;  lanes 0..15 hold x, 16..31 hold y
  const int half = (lane >> 4) & 1;        // 0 = x side, 1 = y side
  const int j0   = (lane & 15) * EPL;      // pair index base
  const float* own = ws + half * ZDIM + j0;
  const float* oth = ws + (half ^ 1) * ZDIM + j0;

  float4 a0 = *(const float4*)(own);
  float4 a1 = *(const float4*)(own + 4);
  float4 b0 = *(const float4*)(oth);
  float4 b1 = *(const float4*)(oth + 4);
  float key[EPL] = {a0.x, a0.y, a0.z, a0.w, a1.x, a1.y, a1.z, a1.w};
  float cv [EPL] = {b0.x, b0.y, b0.z, b0.w, b1.x, b1.y, b1.z, b1.w};
  int   del[EPL];
  const int sgn = half ? 1 : -1;       // y endpoint: +1, x endpoint: -1
#pragma unroll
  for (int r = 0; r < EPL; ++r) {
    // valid pair iff x < y
    bool valid = half ? (cv[r] < key[r]) : (key[r] < cv[r]);
    del[r] = valid ? sgn : 0;
  }

  // ---- blocked bitonic sort of 256 (key,tag) across the wave ----
#pragma unroll
  for (int k = 2; k <= N2; k <<= 1) {
#pragma unroll
    for (int j = N2 / 2; j >= 1; j >>= 1) {
      if (j > k / 2) continue;                      // substages j = k/2 .. 1
      if (j >= EPL) {
        const int lj = j >> 3;                      // partner lane xor mask
#pragma unroll
        for (int r = 0; r < EPL; ++r) {
          const int i = (lane << 3) | r;
          float pk = __shfl_xor(key[r], lj, 32);
          int   pd = __shfl_xor(del[r], lj, 32);
          bool keep_min = ((i & j) == 0) == ((i & k) == 0);
          bool take = keep_min ? (pk < key[r]) : (pk > key[r]);
          key[r] = take ? pk : key[r];
          del[r] = take ? pd : del[r];
        }
      } else {
#pragma unroll
        for (int r = 0; r < EPL; ++r) {
          if ((r & j) == 0) {
            const int q = r | j;
            const int i = (lane << 3) | r;
            bool up = ((i & k) == 0);
            float kp = key[r], kq = key[q];
            int   dp = del[r], dq = del[q];
            bool sw = up ? (kp > kq) : (kp < kq);
            key[r] = sw ? kq : kp;  key[q] = sw ? kp : kq;
            del[r] = sw ? dq : dp;  del[q] = sw ? dp : dq;
          }
        }
      }
    }
  }

  // ---- inclusive scan of tags in sorted order ----
  int cums[EPL];
  int incl = 0;
#pragma unroll
  for (int r = 0; r < EPL; ++r) { incl += del[r]; cums[r] = incl; }
  int scan = incl;                                  // wave inclusive scan
#pragma unroll
  for (int s = 1; s < 32; s <<= 1) {
    int t = __shfl_up(scan, s, 32);
    if (lane >= s) scan += t;
  }
  const int ex = scan - incl;                       // exclusive lane prefix

  // ---- telescoped segment sum:  sum v_i * (f_i - f_{i-1}) ----
  float fprev = (ex < 0) ? -1.0f : 0.0f;            // f of previous element
  float acc = 0.0f;
#pragma unroll
  for (int r = 0; r < EPL; ++r) {
    float fc = ((ex + cums[r]) < 0) ? -1.0f : 0.0f;
    acc += key[r] * (fc - fprev);
    fprev = fc;
  }
#pragma unroll
  for (int s = 16; s >= 1; s >>= 1) acc += __shfl_xor(acc, s, 32);
  if (lane == 0) out[row] = acc;
}

extern "C" void kernel_launch(void* const* d_in, const int* in_sizes, int n_in,
                              void* d_out, int out_size, void* d_ws, size_t ws_size,
                              hipStream_t stream) {
  const float* x = (const float*)d_in[0];
  const float* y = (const float*)d_in[1];
  float* out = (float*)d_out;
  const int nrows = in_sizes[0] / ZDIM;             // 524288
  const int blocks = (nrows + WAVES - 1) / WAVES;   // 65536
  iqe_union_len_kernel<<<blocks, BLK, 0, stream>>>(x, y, out, nrows);
}